// EDLHungarianMatcher_59287728554541
// MI455X (gfx1250) — compile-verified
//
#include <hip/hip_runtime.h>
#include <hip/hip_bf16.h>

typedef __attribute__((ext_vector_type(2))) float v2f;
typedef __attribute__((ext_vector_type(8))) float v8f;
typedef __attribute__((ext_vector_type(4))) int   v4i;

#define B_    4
#define Q_    100
#define T_    20
#define C_    80
#define HW_   16384
#define QT_   7          // ceil(100/16)
#define TT_   2          // ceil(20/16)
#define KCH_  128        // pixels staged in LDS per chunk (32 b128 async ops/chunk)
#define NCH_  8          // chunks processed per wave
#define SW_   16         // K-splits across waves: SW_*NCH_*KCH_ == HW_
#define LSTRIDE_ 132     // KCH_ + 4 pad: 16B-aligned rows, 4-bank skew
#define TGT_OFF_ (16*LSTRIDE_)
#define BUFSZ_   (2*16*LSTRIDE_)   // floats per (out+tgt) buffer

#if __has_builtin(__builtin_amdgcn_global_load_async_to_lds_b128) && \
    __has_builtin(__builtin_amdgcn_s_wait_asynccnt)
#define USE_ASYNC_LDS 1
#else
#define USE_ASYNC_LDS 0
#endif

#define AS1 __attribute__((address_space(1)))
#define AS3 __attribute__((address_space(3)))

// ---------------- softmax stats: one wave32 per (b,q) row ----------------
__global__ __launch_bounds__(32) void softmax_stats_kernel(
    const float* __restrict__ logits, float* __restrict__ smax, float* __restrict__ sden) {
  const int row  = blockIdx.x;            // b*Q + q
  const int lane = threadIdx.x;
  const float* p = logits + (size_t)row * C_;
  float m = -3.4e38f;
  for (int c = lane; c < C_; c += 32) m = fmaxf(m, p[c]);
  for (int off = 16; off > 0; off >>= 1) m = fmaxf(m, __shfl_xor(m, off, 32));
  float s = 0.f;
  for (int c = lane; c < C_; c += 32) s += __expf(p[c] - m);
  for (int off = 16; off > 0; off >>= 1) s += __shfl_xor(s, off, 32);
  if (lane == 0) { smax[row] = m; sden[row] = s; }
}

// ---------------- row sums over HW_ pixels: one block per row ----------------
__global__ __launch_bounds__(256) void rowsum_kernel(
    const float* __restrict__ src, float* __restrict__ dst) {
  __shared__ float sdata[256];
  const int row = blockIdx.x;
  const float* p = src + (size_t)row * HW_;
  float s = 0.f;
  for (int c = threadIdx.x; c < HW_ / 4; c += 256) {
    float4 v = *(const float4*)(p + c * 4);
    s += v.x + v.y + v.z + v.w;
  }
  sdata[threadIdx.x] = s;
  __syncthreads();
  for (int off = 128; off > 0; off >>= 1) {
    if (threadIdx.x < off) sdata[threadIdx.x] += sdata[threadIdx.x + off];
    __syncthreads();
  }
  if (threadIdx.x == 0) dst[row] = sdata[0];
}

// ---- staging helpers: one 16 x KCH_ slab of a [rows x HW_] matrix into LDS ----
__device__ __forceinline__ void stage_tile_sync(
    float* ldsBase, const float* gBase, int row0, int rowMax, int k0, int lane) {
  for (int i = lane; i < 16 * (KCH_ / 4); i += 32) {
    const int r  = i >> 5;            // KCH_/4 == 32
    const int c4 = i & 31;
    int rr = row0 + r; if (rr > rowMax) rr = rowMax;
    const float4 v = *(const float4*)(gBase + (size_t)rr * HW_ + k0 + c4 * 4);
    float* d = ldsBase + r * LSTRIDE_ + c4 * 4;
    d[0] = v.x; d[1] = v.y; d[2] = v.z; d[3] = v.w;
  }
}

#if USE_ASYNC_LDS
__device__ __forceinline__ void stage_tile_async(
    float* ldsBase, const float* gBase, int row0, int rowMax, int k0, int lane) {
  for (int i = lane; i < 16 * (KCH_ / 4); i += 32) {   // 16 async instrs per tile
    const int r  = i >> 5;
    const int c4 = i & 31;
    int rr = row0 + r; if (rr > rowMax) rr = rowMax;
    v4i* g = (v4i*)const_cast<float*>(gBase + (size_t)rr * HW_ + k0 + c4 * 4);
    v4i* l = (v4i*)(ldsBase + r * LSTRIDE_ + c4 * 4);
    __builtin_amdgcn_global_load_async_to_lds_b128(
        (AS1 v4i*)g, (AS3 v4i*)l, /*offset=*/0, /*cpol=*/0);
  }
}
#endif

__device__ __forceinline__ void lds_quiesce() {
#if __has_builtin(__builtin_amdgcn_s_wait_dscnt)
  __builtin_amdgcn_s_wait_dscnt(0);
#else
  __syncthreads();
#endif
  asm volatile("" ::: "memory");
}

// ---------------- main fused tile kernel: WMMA inter + VALU L1 ----------------
// One wave32 per (b, q-tile, t-tile, k-split). Double-buffered async global->LDS
// staging (ASYNCcnt) overlaps fetch with compute; WMMA f32 16x16x4 accumulates
// the dice numerator on the matrix pipe while the VALU loop does pairwise L1.
__global__ __launch_bounds__(32) void pair_tile_kernel(
    const float* __restrict__ out_masks, const float* __restrict__ tgt_masks,
    float* __restrict__ l1_part, float* __restrict__ inter_part) {
  __shared__ float lds[2 * BUFSZ_];   // 33,792 B (double buffer)

  int id = blockIdx.x;
  const int ks = id % SW_; id /= SW_;
  const int tt = id % TT_; id /= TT_;
  const int qt = id % QT_; id /= QT_;
  const int b  = id;

  const int lane = threadIdx.x;
  const int half = lane >> 4;     // 0: lanes 0-15, 1: lanes 16-31
  const int l15  = lane & 15;
  const int q0 = qt * 16, t0 = tt * 16;

  const float* outB = out_masks + (size_t)(b * Q_) * HW_;
  const float* tgtB = tgt_masks + (size_t)(b * T_) * HW_;

  v8f acc = {};                   // WMMA C/D accumulator (inter partial)
  float accL1[8];
  #pragma unroll
  for (int i = 0; i < 8; ++i) accL1[i] = 0.f;

#if USE_ASYNC_LDS
  // Prime buffer 0 with chunk 0 (32 async b128 instrs -> ASYNCcnt += 32).
  {
    const int k0 = ks * KCH_;
    stage_tile_async(lds,            outB, q0, Q_ - 1, k0, lane);
    stage_tile_async(lds + TGT_OFF_, tgtB, t0, T_ - 1, k0, lane);
  }
#endif

  for (int c = 0; c < NCH_; ++c) {
    float* cur = lds + (c & 1) * BUFSZ_;

#if USE_ASYNC_LDS
    if (c + 1 < NCH_) {
      // Buffer (c+1)&1 was last read in the previous iteration; make sure our LDS
      // reads are drained before the async engine overwrites it.
      lds_quiesce();
      float* nxt = lds + ((c + 1) & 1) * BUFSZ_;
      const int k0n = (ks + (c + 1) * SW_) * KCH_;
      stage_tile_async(nxt,            outB, q0, Q_ - 1, k0n, lane);
      stage_tile_async(nxt + TGT_OFF_, tgtB, t0, T_ - 1, k0n, lane);
      // <=32 outstanding => chunk c (issued earlier, completes in order) landed.
      __builtin_amdgcn_s_wait_asynccnt(32);
    } else {
      __builtin_amdgcn_s_wait_asynccnt(0);
    }
    asm volatile("" ::: "memory");
#else
    const int k0 = (ks + c * SW_) * KCH_;
    stage_tile_sync(cur,            outB, q0, Q_ - 1, k0, lane);
    stage_tile_sync(cur + TGT_OFF_, tgtB, t0, T_ - 1, k0, lane);
    __syncthreads();
#endif

    // ---- WMMA: inter += outTile(16xK) x tgtTile^T(Kx16), K-step 4 ----
    // A 16x4 layout: lane<16 -> (M=lane, K=0,1), lane>=16 -> (M=lane-16, K=2,3)
    const float* aRow = cur + l15 * LSTRIDE_;
    const float* bRow = cur + TGT_OFF_ + l15 * LSTRIDE_;
    const int kofs = half * 2;
    for (int kk = 0; kk < KCH_; kk += 4) {
      v2f a  = *(const v2f*)(aRow + kk + kofs);
      v2f bv = *(const v2f*)(bRow + kk + kofs);
      acc = __builtin_amdgcn_wmma_f32_16x16x4_f32(
          /*neg_a=*/false, a, /*neg_b=*/false, bv,
          /*c_mod=*/(short)0, acc, /*reuse_a=*/false, /*reuse_b=*/false);
    }

    // ---- VALU: pairwise L1 on the same C-matrix lane mapping ----
    // lane owns N = l15, M = half*8 + i  (matches acc[i] pair assignment)
    const float* tRow  = bRow;
    const float* oBase = cur + (half * 8) * LSTRIDE_;
    for (int k = 0; k < KCH_; k += 2) {
      v2f tv = *(const v2f*)(tRow + k);
      #pragma unroll
      for (int i = 0; i < 8; ++i) {
        v2f ov = *(const v2f*)(oBase + i * LSTRIDE_ + k);
        accL1[i] += __builtin_fabsf(ov.x - tv.x) + __builtin_fabsf(ov.y - tv.y);
      }
    }

#if !USE_ASYNC_LDS
    __syncthreads();
#endif
  }

  // ---- deterministic partial stores (no atomics): slot per (b,q,t,ks) ----
  const int t = t0 + l15;
  #pragma unroll
  for (int i = 0; i < 8; ++i) {
    const int q = q0 + half * 8 + i;
    if (q < Q_ && t < T_) {
      const size_t slot = (((size_t)(b * Q_ + q)) * T_ + t) * SW_ + ks;
      inter_part[slot] = acc[i];
      l1_part[slot]    = accL1[i];
    }
  }
}

// ---------------- combine: sum K-partials, fuse class + dice terms ----------------
__global__ __launch_bounds__(256) void combine_kernel(
    const float* __restrict__ out_probs, const int* __restrict__ labels,
    const float* __restrict__ l1_part, const float* __restrict__ inter_part,
    const float* __restrict__ osum, const float* __restrict__ tsum,
    const float* __restrict__ smax, const float* __restrict__ sden,
    float* __restrict__ out) {
  const int idx = blockIdx.x * blockDim.x + threadIdx.x;
  if (idx >= B_ * Q_ * T_) return;
  const int t  = idx % T_;
  const int bq = idx / T_;            // b*Q + q
  const int b  = bq / Q_;

  float l1 = 0.f, inter = 0.f;
  const float* lp = l1_part    + (size_t)idx * SW_;
  const float* ip = inter_part + (size_t)idx * SW_;
  #pragma unroll
  for (int s = 0; s < SW_; ++s) { l1 += lp[s]; inter += ip[s]; }

  const int lab     = labels[b * T_ + t];
  const float logit = out_probs[(size_t)bq * C_ + lab];
  const float cls   = -__expf(logit - smax[bq]) / sden[bq];
  const float dice  = 1.f - (2.f * inter + 1.f) / (osum[bq] + tsum[b * T_ + t] + 1.f);
  out[idx] = l1 + cls + dice;
}

extern "C" void kernel_launch(void* const* d_in, const int* in_sizes, int n_in,
                              void* d_out, int out_size, void* d_ws, size_t ws_size,
                              hipStream_t stream) {
  (void)in_sizes; (void)n_in; (void)out_size; (void)ws_size;
  const float* out_probs = (const float*)d_in[0];
  const float* out_masks = (const float*)d_in[1];
  const float* tgt_masks = (const float*)d_in[2];
  const int*   labels    = (const int*)d_in[3];

  float* ws        = (float*)d_ws;
  float* l1_part   = ws;                      // B*Q*T*SW_ = 128000
  float* inter_part= ws + 128000;             // 128000
  float* osum      = ws + 256000;             // 400
  float* tsum      = ws + 256400;             // 80
  float* smax      = ws + 256480;             // 400
  float* sden      = ws + 256880;             // 400   (~1.03 MB total)

  softmax_stats_kernel<<<B_ * Q_, 32, 0, stream>>>(out_probs, smax, sden);
  rowsum_kernel<<<B_ * Q_, 256, 0, stream>>>(out_masks, osum);
  rowsum_kernel<<<B_ * T_, 256, 0, stream>>>(tgt_masks, tsum);
  pair_tile_kernel<<<B_ * QT_ * TT_ * SW_, 32, 0, stream>>>(out_masks, tgt_masks,
                                                            l1_part, inter_part);
  combine_kernel<<<(B_ * Q_ * T_ + 255) / 256, 256, 0, stream>>>(
      out_probs, labels, l1_part, inter_part, osum, tsum, smax, sden, (float*)d_out);
}